// QDDPM_5557687681068
// MI455X (gfx1250) — compile-verified
//
#include <hip/hip_runtime.h>
#include <math.h>

typedef __attribute__((ext_vector_type(2))) float v2f;
typedef __attribute__((ext_vector_type(8))) float v8f;

#define QTOT   13
#define DN     512
#define DTOT   8192
#define LAYERS 4
#define TSTEPS 20
#define BATCH  128
#define PROW   (2*QTOT*LAYERS)   /* 104 params per step */

__device__ __forceinline__ float2 cmul(float2 a, float2 b) {
  return make_float2(a.x*b.x - a.y*b.y, a.x*b.y + a.y*b.x);
}

__global__ __launch_bounds__(256, 1)
void qddpm_kernel(const float* __restrict__ xre, const float* __restrict__ xim,
                  const float* __restrict__ params, float* __restrict__ out)
{
  // LDS-resident statevector: one batch element per workgroup.
  __shared__ float2 sbuf0[DTOT];      // 64 KB
  __shared__ float2 sbuf1[DTOT];      // 64 KB (ping-pong)
  __shared__ float2 Umat[4][256];     // 4 group unitaries, 16x16 complex each
  __shared__ float2 gate[QTOT + 1][4];// fused RY*RX per qubit (+identity slot)
  __shared__ float  probs[16];
  __shared__ float  s_ninv;
  __shared__ int    s_sel;

  const int b    = blockIdx.x;
  const int tid  = threadIdx.x;
  const int lane = tid & 31;
  const int wave = tid >> 5;
  const int row  = lane & 15;   // WMMA M-row / N-col for this lane
  const int hi   = lane >> 4;   // half-wave selector in f32 WMMA layouts

  // ---- load input into LDS (ancillas |0..0>) and emit slot T (init) ----
  {
    float* o = out + (size_t)(TSTEPS * BATCH + b) * DTOT * 2;
    for (int i = tid; i < DTOT; i += 256) {
      float re = 0.f, im = 0.f;
      if (i < DN) { re = xre[b*DN + i]; im = xim[b*DN + i]; }
      sbuf0[i] = make_float2(re, im);
      __builtin_nontemporal_store(re, o + 2*i);
      __builtin_nontemporal_store(im, o + 2*i + 1);
    }
  }
  __syncthreads();

  float2* cur = sbuf0;
  float2* nxt = sbuf1;

  for (int j = 0; j < TSTEPS; ++j) {
    const float* prow = params + (size_t)(TSTEPS - 1 - j) * PROW;
    if (j + 1 < TSTEPS)
      __builtin_prefetch(params + (size_t)(TSTEPS - 2 - j) * PROW, 0, 1);

    for (int l = 0; l < LAYERS; ++l) {
      // ---- fused per-qubit gate G = RY(ty)*RX(tx) ----
      if (tid < QTOT) {
        float ha = 0.5f * prow[2*QTOT*l + tid];
        float hb = 0.5f * prow[2*QTOT*l + QTOT + tid];
        float sa, ca, sb, cb;
        __sincosf(ha, &sa, &ca);
        __sincosf(hb, &sb, &cb);
        gate[tid][0] = make_float2( cb*ca,  sb*sa);
        gate[tid][1] = make_float2(-sb*ca, -cb*sa);
        gate[tid][2] = make_float2( sb*ca, -cb*sa);
        gate[tid][3] = make_float2( cb*ca, -sb*sa);
      } else if (tid == QTOT) {
        gate[QTOT][0] = make_float2(1.f, 0.f);
        gate[QTOT][1] = make_float2(0.f, 0.f);
        gate[QTOT][2] = make_float2(0.f, 0.f);
        gate[QTOT][3] = make_float2(1.f, 0.f);
      }
      __syncthreads();

      // ---- build 16x16 Kronecker-group unitaries (one entry per thread/group) ----
      {
        const int s = tid >> 4, t = tid & 15;
        #pragma unroll
        for (int g = 0; g < 4; ++g) {
          const int q0 = (g < 3) ? 4*g     : QTOT;   // group 3 = I,I,I,G12 on bits 3..0
          const int q1 = (g < 3) ? 4*g + 1 : QTOT;
          const int q2 = (g < 3) ? 4*g + 2 : QTOT;
          const int q3 = (g < 3) ? 4*g + 3 : 12;
          float2 e0 = gate[q0][(((s>>3)&1)<<1) | ((t>>3)&1)];
          float2 e1 = gate[q1][(((s>>2)&1)<<1) | ((t>>2)&1)];
          float2 e2 = gate[q2][(((s>>1)&1)<<1) | ((t>>1)&1)];
          float2 e3 = gate[q3][(((s   )&1)<<1) | ((t   )&1)];
          Umat[g][tid] = cmul(cmul(e0, e1), cmul(e2, e3));
        }
      }
      __syncthreads();

      // ---- apply the 4 group unitaries: 16x16 complex GEMM via f32 WMMA ----
      #pragma unroll
      for (int g = 0; g < 4; ++g) {
        const int psh   = (g==0) ? 9 : (g==1) ? 5 : (g==2) ? 1 : 0;
        const int postN = 1 << psh;

        // A fragments (U, 16x4 per K-chunk): VGPR0 = K {0,2}, VGPR1 = K {1,3}
        v2f ur[4], ui[4], nui[4];
        #pragma unroll
        for (int k = 0; k < 4; ++k) {
          float2 u0 = Umat[g][row*16 + 4*k + 2*hi];
          float2 u1 = Umat[g][row*16 + 4*k + 2*hi + 1];
          v2f a; a.x = u0.x; a.y = u1.x; ur[k]  = a;
          v2f c; c.x = u0.y; c.y = u1.y; ui[k]  = c;
          v2f n; n.x = -u0.y; n.y = -u1.y; nui[k] = n;
        }

        for (int ct = wave; ct < 32; ct += 8) {       // 32 column tiles / 8 waves
          const int col  = ct*16 + row;
          const int base = ((col >> psh) << (psh + 4)) + (col & (postN - 1));
          v8f acc_re = {}; v8f acc_im = {};
          #pragma unroll
          for (int k = 0; k < 4; ++k) {
            const int ta = 4*k + 2*hi;
            float2 e0 = cur[base + ta*postN];
            float2 e1 = cur[base + (ta+1)*postN];
            v2f sr; sr.x = e0.x; sr.y = e1.x;
            v2f si; si.x = e0.y; si.y = e1.y;
            // D_re += Ur*Sr - Ui*Si ; D_im += Ur*Si + Ui*Sr
            acc_re = __builtin_amdgcn_wmma_f32_16x16x4_f32(false, ur[k],  false, sr, (short)0, acc_re, false, false);
            acc_re = __builtin_amdgcn_wmma_f32_16x16x4_f32(false, nui[k], false, si, (short)0, acc_re, false, false);
            acc_im = __builtin_amdgcn_wmma_f32_16x16x4_f32(false, ur[k],  false, si, (short)0, acc_im, false, false);
            acc_im = __builtin_amdgcn_wmma_f32_16x16x4_f32(false, ui[k],  false, sr, (short)0, acc_im, false, false);
          }
          #pragma unroll
          for (int r = 0; r < 8; ++r) {
            const int orow = r + 8*hi;
            nxt[base + orow*postN] = make_float2(acc_re[r], acc_im[r]);
          }
        }
        __syncthreads();
        float2* tmp = cur; cur = nxt; nxt = tmp;
      }

      // ---- fused CZ chains: sign = parity of adjacent-bit AND pairs ----
      for (int i = tid; i < DTOT; i += 256) {
        if (__popc(i & (i >> 1)) & 1) {
          float2 v = cur[i];
          cur[i] = make_float2(-v.x, -v.y);
        }
      }
      __syncthreads();
    } // layers

    // ---- measure 4 ancillas (top 4 index bits): probs over 16 blocks of 512 ----
    if (tid < 16) probs[tid] = 0.f;
    __syncthreads();
    {
      float acc = 0.f;
      const int basei = tid * 32;        // 32 contiguous amps, all in block tid>>4
      #pragma unroll 4
      for (int i = 0; i < 32; ++i) {
        float2 v = cur[basei + i];
        acc += v.x*v.x + v.y*v.y;
      }
      atomicAdd(&probs[tid >> 4], acc);
    }
    __syncthreads();
    if (tid == 0) {
      // deterministic counter-based surrogate for jax.random.categorical
      unsigned int h = 0x9E3779B9u ^ (1664525u*(unsigned)j + 1013904223u*(unsigned)b + 42u);
      h ^= h >> 16; h *= 0x85EBCA6Bu; h ^= h >> 13; h *= 0xC2B2AE35u; h ^= h >> 16;
      const float u = (float)(h >> 8) * (1.0f / 16777216.0f);
      float tot = 0.f;
      #pragma unroll
      for (int m = 0; m < 16; ++m) tot += probs[m];
      const float target = u * tot;
      float cdf = 0.f; int sel = 15; bool done = false;
      #pragma unroll
      for (int m = 0; m < 16; ++m) {
        cdf += probs[m];
        if (!done && cdf >= target) { sel = m; done = true; }
      }
      s_sel  = sel;
      s_ninv = rsqrtf(fmaxf(probs[sel], 1e-20f));
    }
    __syncthreads();
    const int   sel  = s_sel;
    const float ninv = s_ninv;

    // ---- collapse + normalize; stream this step's padded state to HBM ----
    float* o = out + (size_t)((TSTEPS - 1 - j) * BATCH + b) * DTOT * 2;
    for (int i = tid; i < DN; i += 256) {
      float2 s = cur[sel*DN + i];
      float2 v = make_float2(s.x*ninv, s.y*ninv);
      nxt[i] = v;                                   // stage in the spare buffer
      __builtin_nontemporal_store(v.x, o + 2*i);
      __builtin_nontemporal_store(v.y, o + 2*i + 1);
    }
    for (int i = DN + tid; i < DTOT; i += 256) {
      __builtin_nontemporal_store(0.f, o + 2*i);
      __builtin_nontemporal_store(0.f, o + 2*i + 1);
    }
    __syncthreads();
    for (int i = tid; i < DTOT; i += 256)
      cur[i] = (i < DN) ? nxt[i] : make_float2(0.f, 0.f);
    __syncthreads();
  } // steps
}

extern "C" void kernel_launch(void* const* d_in, const int* in_sizes, int n_in,
                              void* d_out, int out_size, void* d_ws, size_t ws_size,
                              hipStream_t stream) {
  const float* xre    = (const float*)d_in[0];  // (128, 512) f32
  const float* xim    = (const float*)d_in[1];  // (128, 512) f32
  const float* params = (const float*)d_in[2];  // (20, 104)  f32
  float* out = (float*)d_out;                   // (21,128,8192) complex64 interleaved
  (void)in_sizes; (void)n_in; (void)out_size; (void)d_ws; (void)ws_size;
  qddpm_kernel<<<dim3(BATCH), dim3(256), 0, stream>>>(xre, xim, params, out);
}